// ResnetBottleneckBlock_90718299226283
// MI455X (gfx1250) — compile-verified
//
#include <hip/hip_runtime.h>

// ---------------------------------------------------------------------------
// KPConv ResNet bottleneck block for MI455X (gfx1250), fp32 WMMA + TDM.
//   N=50000 pts, H=32 neighbors, K=15 kernel pts, IN=256, MID=128, OUT=512.
// Matrix math:  V_WMMA_F32_16X16X4_F32 (wave32).
// Data movement: TENSOR_LOAD_TO_LDS (D# per ISA ch.8), double buffered,
//                synced with s_wait_tensorcnt + workgroup barriers.
// ---------------------------------------------------------------------------

#define NPTS_IN   256
#define NPTS_MID  128
#define NPTS_OUT  512
#define NBH       32
#define KPK       15
#define EXTENT    1.2f
#define BN_EPS    1e-5f

typedef __attribute__((ext_vector_type(2))) float        v2f;
typedef __attribute__((ext_vector_type(8))) float        v8f;
typedef __attribute__((ext_vector_type(4))) unsigned int u32x4;
typedef __attribute__((ext_vector_type(4))) int          i32x4;
typedef __attribute__((ext_vector_type(8))) int          i32x8;

__device__ __forceinline__ v8f wmma4(v2f a, v2f b, v8f c) {
  return __builtin_amdgcn_wmma_f32_16x16x4_f32(false, a, false, b, (short)0, c,
                                               false, false);
}

__device__ __forceinline__ unsigned lds_off(const void* p) {
  // Generic address of an LDS object: low 32 bits are the LDS byte offset.
  return (unsigned)(unsigned long long)(uintptr_t)p;
}

// Tensor DMA: load a 2D fp32 tile [tile_d1 rows x tile_d0 elems] from global
// (row stride row_stride elems) into LDS, with optional per-row LDS padding
// (pad codes per ISA: interval 0..7 => 2,4,8,...,256 DW; amount n => n+1 DW).
// Tensor dims bound OOB reads (rows past tens_d1 return zero).
__device__ __forceinline__ void tdm_load_2d(unsigned lds_byte,
                                            const float* gaddr,
                                            unsigned tile_d0, unsigned tile_d1,
                                            unsigned tens_d0, unsigned tens_d1,
                                            unsigned row_stride,
                                            int pad_en, unsigned pad_int_code,
                                            unsigned pad_amt_code) {
  unsigned long long ga = (unsigned long long)(uintptr_t)gaddr;
  u32x4 g0;
  g0[0] = 1u;                                              // count=1 (valid D#)
  g0[1] = lds_byte;                                        // lds_addr
  g0[2] = (unsigned)ga;                                    // global_addr[31:0]
  g0[3] = (unsigned)((ga >> 32) & 0x1FFFFFFull) | (2u << 30); // addr[56:32]|type=2
  unsigned w0 = (2u << 16)                                 // data_size = 4B
              | ((unsigned)(pad_en & 1) << 20)
              | (pad_int_code << 22) | (pad_amt_code << 25);
  unsigned long long q0 = (unsigned long long)w0
                        | ((unsigned long long)(tens_d0 & 0xFFFFu) << 48);
  unsigned long long q1 = ((unsigned long long)(tens_d0 >> 16) & 0xFFFFull)
                        | (((unsigned long long)tens_d1 & 0xFFFFFFFFull) << 16)
                        | ((unsigned long long)(tile_d0 & 0xFFFFu) << 48);
  unsigned long long q2 = (unsigned long long)(tile_d1 & 0xFFFFu)   // tile_dim2=0
                        | ((unsigned long long)row_stride << 32);   // dim0 stride
  unsigned long long q3 = 0ull;                            // stride hi / dim1 stride
  i32x8 g1;
  g1[0] = (int)(unsigned)q0; g1[1] = (int)(q0 >> 32);
  g1[2] = (int)(unsigned)q1; g1[3] = (int)(q1 >> 32);
  g1[4] = (int)(unsigned)q2; g1[5] = (int)(q2 >> 32);
  g1[6] = (int)(unsigned)q3; g1[7] = (int)(q3 >> 32);
  i32x4 z4 = (i32x4){0, 0, 0, 0};
#if defined(__clang_major__) && (__clang_major__ >= 23)
  i32x8 z8 = (i32x8){0, 0, 0, 0, 0, 0, 0, 0};
  __builtin_amdgcn_tensor_load_to_lds(g0, g1, z4, z4, z8, 0);
#else
  __builtin_amdgcn_tensor_load_to_lds(g0, g1, z4, z4, 0);
#endif
}

// ---------------------------------------------------------------------------
// Tiled fp32 GEMM: C[M,N] = A[M,K] * B[K,N], row-major.
// Block 256 thr (8 waves) -> 128(M) x 64(N) C tile, K in 32-wide blocks.
// TDM double-buffers A (128x32, LDS stride 34) and B (32x64, LDS stride 72);
// wave 0 drives the DMA, one barrier per K block. Each wave: 32x32 C tile.
// ---------------------------------------------------------------------------
#define GA_STRIDE 34   // 32 + 2 DW pad  (interval code 4, amount code 1)
#define GB_STRIDE 72   // 64 + 8 DW pad  (interval code 5, amount code 7)

__global__ __launch_bounds__(256)
void k_gemm_f32(const float* __restrict__ A, const float* __restrict__ B,
                float* __restrict__ C, int M, int N, int Kd) {
  extern __shared__ float ls[];
  float* As = ls;                          // [2][128][GA_STRIDE]
  float* Bs = ls + 2 * 128 * GA_STRIDE;    // [2][32][GB_STRIDE]

  const int tid = threadIdx.x, lane = tid & 31, wave = tid >> 5;
  const int half = lane >> 4, l16 = lane & 15;
  const int m0 = blockIdx.x * 128, n0 = blockIdx.y * 64;
  const int mW = (wave >> 1) * 32, nW = (wave & 1) * 32;
  const int nb = Kd >> 5;
  int mrem = M - m0; if (mrem > 128) mrem = 128;

  v8f acc[2][2];
#pragma unroll
  for (int i = 0; i < 2; ++i)
#pragma unroll
    for (int j = 0; j < 2; ++j) acc[i][j] = (v8f){0,0,0,0,0,0,0,0};

  if (tid == 0) {
    tdm_load_2d(lds_off(As), A + (size_t)m0 * Kd, 32, 128, 32, (unsigned)mrem,
                (unsigned)Kd, 1, 4, 1);
    tdm_load_2d(lds_off(Bs), B + n0, 64, 32, 64, 32, (unsigned)N, 1, 5, 7);
  }

  for (int kb = 0; kb < nb; ++kb) {
    if (tid == 0) __builtin_amdgcn_s_wait_tensorcnt(0);
    __syncthreads();   // buffers for kb ready; all waves done with kb-1
    if (tid == 0 && kb + 1 < nb) {
      int nxt = (kb + 1) & 1;
      tdm_load_2d(lds_off(As + nxt * 128 * GA_STRIDE),
                  A + (size_t)m0 * Kd + (kb + 1) * 32, 32, 128, 32,
                  (unsigned)mrem, (unsigned)Kd, 1, 4, 1);
      tdm_load_2d(lds_off(Bs + nxt * 32 * GB_STRIDE),
                  B + (size_t)(kb + 1) * 32 * N + n0, 64, 32, 64, 32,
                  (unsigned)N, 1, 5, 7);
    }
    const float* Ab = As + (kb & 1) * 128 * GA_STRIDE;
    const float* Bb = Bs + (kb & 1) * 32 * GB_STRIDE;
#pragma unroll
    for (int ks = 0; ks < 32; ks += 4) {
      v2f a0 = *(const v2f*)(Ab + (mW + l16) * GA_STRIDE + ks + 2 * half);
      v2f a1 = *(const v2f*)(Ab + (mW + 16 + l16) * GA_STRIDE + ks + 2 * half);
#pragma unroll
      for (int t = 0; t < 2; ++t) {
        const float* bp = Bb + (ks + 2 * half) * GB_STRIDE + nW + t * 16 + l16;
        v2f b; b.x = bp[0]; b.y = bp[GB_STRIDE];
        acc[0][t] = wmma4(a0, b, acc[0][t]);
        acc[1][t] = wmma4(a1, b, acc[1][t]);
      }
    }
  }

#pragma unroll
  for (int mt = 0; mt < 2; ++mt)
#pragma unroll
    for (int t = 0; t < 2; ++t)
#pragma unroll
      for (int j = 0; j < 8; ++j) {
        int m = m0 + mW + mt * 16 + j + half * 8;
        if (m < M) C[(size_t)m * N + n0 + nW + t * 16 + l16] = acc[mt][t][j];
      }
}

// ---------------------------------------------------------------------------
// BN stats / normalize / combine (unchanged logic).
// ---------------------------------------------------------------------------
__global__ __launch_bounds__(128)
void k_colstats(const float* __restrict__ X, int M, int C, float* __restrict__ st) {
  int c = blockIdx.x * 128 + threadIdx.x;
  if (c >= C) return;
  int chunk = (M + gridDim.y - 1) / gridDim.y;
  int r0 = blockIdx.y * chunk;
  int r1 = r0 + chunk; if (r1 > M) r1 = M;
  float s = 0.f, q = 0.f;
  for (int r = r0; r < r1; ++r) {
    float v = X[(size_t)r * C + c];
    s += v; q += v * v;
  }
  atomicAdd(&st[c], s);
  atomicAdd(&st[C + c], q);
}

__global__ __launch_bounds__(256)
void k_bn_act(float* __restrict__ X, const float* __restrict__ st,
              const float* __restrict__ g, const float* __restrict__ b,
              int M, int C, int act) {
  size_t total = (size_t)M * C;
  float invM = 1.0f / (float)M;
  for (size_t i = (size_t)blockIdx.x * blockDim.x + threadIdx.x; i < total;
       i += (size_t)gridDim.x * blockDim.x) {
    int c = (int)(i & (size_t)(C - 1));
    float mu  = st[c] * invM;
    float var = st[C + c] * invM - mu * mu;
    float y = g[c] * (X[i] - mu) * rsqrtf(var + BN_EPS) + b[c];
    if (act) y = (y >= 0.f) ? y : 0.1f * y;
    X[i] = y;
  }
}

__global__ __launch_bounds__(256)
void k_combine(const float* __restrict__ u2, float* __restrict__ out,
               const float* __restrict__ stu, const float* __restrict__ sts,
               const float* __restrict__ g2, const float* __restrict__ b2,
               const float* __restrict__ gs, const float* __restrict__ bs,
               int M, int C) {
  size_t total = (size_t)M * C;
  float invM = 1.0f / (float)M;
  for (size_t i = (size_t)blockIdx.x * blockDim.x + threadIdx.x; i < total;
       i += (size_t)gridDim.x * blockDim.x) {
    int c = (int)(i & (size_t)(C - 1));
    float muu = stu[c] * invM, varu = stu[C + c] * invM - muu * muu;
    float mus = sts[c] * invM, vars = sts[C + c] * invM - mus * mus;
    float yu = g2[c] * (u2[i]  - muu) * rsqrtf(varu + BN_EPS) + b2[c];
    float ys = gs[c] * (out[i] - mus) * rsqrtf(vars + BN_EPS) + bs[c];
    float y = yu + ys;
    out[i] = (y >= 0.f) ? y : 0.1f * y;
  }
}

// ---------------------------------------------------------------------------
// Fused KPConv: one 256-thr workgroup handles 16 points.
//  Phase 0: influence weights w[p][k][h] -> LDS (stride 34; row k=15 zero).
//  Phase 1: weighted[p] = w[p]^T(16x32) @ x1[nb](32x128); WMMA, gathered B.
//           D stored to LDS as wt[kk][p] (kk=k*128+c, point stride 17).
//  Phase 2: out(16x128) = WtFlat(16x1920) @ WkFlat(1920x128); WMMA with Wk
//           streamed through TDM double buffers (32x128 tiles, LDS stride 136),
//           first DMA issued at kernel entry to overlap phases 0-1.
// LDS total ~198 KB of the 320 KB WGP LDS.
// ---------------------------------------------------------------------------
#define WB_STRIDE 34
#define WT_PSTR   17
#define WK_STRIDE 136  // 128 + 8 DW pad (interval code 6, amount code 7)
#define KP_WBUF_F (16 * 16 * WB_STRIDE)          // 8704 floats
#define KP_WT_F   (KPK * 128 * WT_PSTR)          // 32640 floats
#define KP_WK_F   (2 * 32 * WK_STRIDE)           // 8704 floats
#define KP_STEPS  (KPK * 4)                      // 60 c-blocks of 32

__global__ __launch_bounds__(256)
void k_kpconv(const float* __restrict__ x1, const float* __restrict__ pts,
              const int* __restrict__ nb, const float* __restrict__ Kp,
              const float* __restrict__ Wk, float* __restrict__ out, int Npts) {
  extern __shared__ float smem[];
  float* wbuf  = smem;                            // [16][16][WB_STRIDE]
  float* wt    = smem + KP_WBUF_F;                // [KPK*128][WT_PSTR]
  float* wkbuf = wt + KP_WT_F;                    // [2][32][WK_STRIDE]
  int*   idx   = (int*)(wkbuf + KP_WK_F);         // [16][32]

  const int tid  = threadIdx.x;
  const int lane = tid & 31, wave = tid >> 5;
  const int half = lane >> 4, l16 = lane & 15;
  const int p0   = blockIdx.x * 16;

  // Kick off the first Wk tile DMA immediately (overlaps phases 0-1).
  if (tid == 0)
    tdm_load_2d(lds_off(wkbuf), Wk, 128, 32, 128, 32, 128, 1, 6, 7);

  // ---- Phase 0: influence weights ----
  for (int i = tid; i < 16 * NBH; i += 256) {
    int p = i >> 5, h = i & (NBH - 1);
    int n = p0 + p; if (n >= Npts) n = Npts - 1;
    int j = nb[(size_t)n * NBH + h];
    idx[p * NBH + h] = j;
    float rx = pts[(size_t)j * 3 + 0] - pts[(size_t)n * 3 + 0];
    float ry = pts[(size_t)j * 3 + 1] - pts[(size_t)n * 3 + 1];
    float rz = pts[(size_t)j * 3 + 2] - pts[(size_t)n * 3 + 2];
#pragma unroll
    for (int k = 0; k < KPK; ++k) {
      float dx = rx - Kp[k * 3 + 0];
      float dy = ry - Kp[k * 3 + 1];
      float dz = rz - Kp[k * 3 + 2];
      float d  = sqrtf(dx * dx + dy * dy + dz * dz);
      wbuf[(p * 16 + k) * WB_STRIDE + h] = fmaxf(1.f - d * (1.f / EXTENT), 0.f);
    }
    wbuf[(p * 16 + 15) * WB_STRIDE + h] = 0.f;     // pad row
  }
  __syncthreads();

  // ---- Phase 1: weighted = w^T @ x1[nb] ---- (16 points x 8 c-tiles)
  for (int job = wave; job < 128; job += 8) {
    int p = job >> 3, ct = job & 7;
    const float* aBase = &wbuf[(p * 16 + l16) * WB_STRIDE + 2 * half];
    const int*   ip    = &idx[p * NBH];
    v8f acc = (v8f){0,0,0,0,0,0,0,0};
#pragma unroll
    for (int h = 0; h < NBH; h += 4) {
      v2f a = *(const v2f*)(aBase + h);
      int r0 = ip[h + 2 * half + 0];
      int r1 = ip[h + 2 * half + 1];
      v2f b;
      b.x = x1[(size_t)r0 * NPTS_MID + ct * 16 + l16];
      b.y = x1[(size_t)r1 * NPTS_MID + ct * 16 + l16];
      acc = wmma4(a, b, acc);
    }
#pragma unroll
    for (int j = 0; j < 8; ++j) {
      int mrow = j + half * 8;                     // kernel-point row
      if (mrow < KPK)
        wt[(mrow * 128 + ct * 16 + l16) * WT_PSTR + p] = acc[j];
    }
  }
  __syncthreads();

  // ---- Phase 2: out = WtFlat @ WkFlat, Wk via TDM double buffer ----
  {
    const int dt = wave;                           // 8 waves x 16 out-cols
    v8f acc = (v8f){0,0,0,0,0,0,0,0};
    for (int s = 0; s < KP_STEPS; ++s) {
      if (tid == 0) __builtin_amdgcn_s_wait_tensorcnt(0);
      __syncthreads();
      if (tid == 0 && s + 1 < KP_STEPS) {
        int kn = (s + 1) >> 2, cbn = (s + 1) & 3;
        tdm_load_2d(lds_off(wkbuf + ((s + 1) & 1) * 32 * WK_STRIDE),
                    Wk + ((size_t)kn * 128 + cbn * 32) * 128,
                    128, 32, 128, 32, 128, 1, 6, 7);
      }
      const int k = s >> 2, cb = s & 3;
      const float* wb = wkbuf + (s & 1) * 32 * WK_STRIDE;
      const int kkBase = k * 128 + cb * 32;
#pragma unroll
      for (int c4 = 0; c4 < 32; c4 += 4) {
        int kk = kkBase + c4 + 2 * half;
        v2f a;
        a.x = wt[(size_t)kk * WT_PSTR + l16];
        a.y = wt[(size_t)(kk + 1) * WT_PSTR + l16];
        v2f b;
        b.x = wb[(c4 + 2 * half) * WK_STRIDE + dt * 16 + l16];
        b.y = wb[(c4 + 2 * half + 1) * WK_STRIDE + dt * 16 + l16];
        acc = wmma4(a, b, acc);
      }
      __syncthreads();                             // done with this buffer
    }
#pragma unroll
    for (int j = 0; j < 8; ++j) {
      int n = p0 + j + half * 8;
      if (n < Npts) out[(size_t)n * NPTS_MID + dt * 16 + l16] = acc[j];
    }
  }
}

// ---------------------------------------------------------------------------
extern "C" void kernel_launch(void* const* d_in, const int* in_sizes, int n_in,
                              void* d_out, int out_size, void* d_ws, size_t ws_size,
                              hipStream_t stream) {
  const float* features = (const float*)d_in[0];
  const float* pts      = (const float*)d_in[1];
  const int*   nbi      = (const int*)  d_in[2];
  const float* W1 = (const float*)d_in[3];
  const float* g1 = (const float*)d_in[4];
  const float* b1 = (const float*)d_in[5];
  const float* Kp = (const float*)d_in[6];
  const float* Wk = (const float*)d_in[7];
  const float* gk = (const float*)d_in[8];
  const float* bk = (const float*)d_in[9];
  const float* W2 = (const float*)d_in[10];
  const float* g2 = (const float*)d_in[11];
  const float* b2 = (const float*)d_in[12];
  const float* Ws = (const float*)d_in[13];
  const float* gs = (const float*)d_in[14];
  const float* bs = (const float*)d_in[15];
  float* outp = (float*)d_out;

  const int N = in_sizes[0] / NPTS_IN;   // 50000

  float* wsf = (float*)d_ws;
  float* st1 = wsf;                 // 2*128
  float* stk = wsf + 256;           // 2*128
  float* stu = wsf + 512;           // 2*512
  float* sts = wsf + 1536;          // 2*512
  float* x1  = wsf + 2560;                         // [N,128]
  float* kp  = x1 + (size_t)N * NPTS_MID;          // [N,128]
  float* u2  = kp + (size_t)N * NPTS_MID;          // [N,512]

  hipMemsetAsync(wsf, 0, 2560 * sizeof(float), stream);

  const size_t gemmLds = (size_t)(2 * 128 * GA_STRIDE + 2 * 32 * GB_STRIDE)
                       * sizeof(float);            // 53,248 B
  const size_t kpLds   = (size_t)(KP_WBUF_F + KP_WT_F + KP_WK_F) * sizeof(float)
                       + 16 * NBH * sizeof(int);   // ~202 KB

  // --- unary1: x1 = features @ W1; BN + LeakyReLU ---
  k_gemm_f32<<<dim3((N + 127) / 128, NPTS_MID / 64), 256, gemmLds, stream>>>(
      features, W1, x1, N, NPTS_MID, NPTS_IN);
  k_colstats<<<dim3(1, 64), 128, 0, stream>>>(x1, N, NPTS_MID, st1);
  k_bn_act<<<2048, 256, 0, stream>>>(x1, st1, g1, b1, N, NPTS_MID, 1);

  // --- fused KPConv -> kp; BN + LeakyReLU ---
  k_kpconv<<<(N + 15) / 16, 256, kpLds, stream>>>(x1, pts, nbi, Kp, Wk, kp, N);
  k_colstats<<<dim3(1, 64), 128, 0, stream>>>(kp, N, NPTS_MID, stk);
  k_bn_act<<<2048, 256, 0, stream>>>(kp, stk, gk, bk, N, NPTS_MID, 1);

  // --- unary2: u2 = kp @ W2 ---
  k_gemm_f32<<<dim3((N + 127) / 128, NPTS_OUT / 64), 256, gemmLds, stream>>>(
      kp, W2, u2, N, NPTS_OUT, NPTS_MID);
  // --- shortcut: d_out = features @ Ws ---
  k_gemm_f32<<<dim3((N + 127) / 128, NPTS_OUT / 64), 256, gemmLds, stream>>>(
      features, Ws, outp, N, NPTS_OUT, NPTS_IN);

  k_colstats<<<dim3(4, 64), 128, 0, stream>>>(u2,   N, NPTS_OUT, stu);
  k_colstats<<<dim3(4, 64), 128, 0, stream>>>(outp, N, NPTS_OUT, sts);

  // --- out = lrelu(bn(u2) + bn(shortcut)) ---
  k_combine<<<4096, 256, 0, stream>>>(u2, outp, stu, sts, g2, b2, gs, bs,
                                      N, NPTS_OUT);
}